// CapsuleLayer_69002944578111
// MI455X (gfx1250) — compile-verified
//
#include <hip/hip_runtime.h>

typedef __attribute__((ext_vector_type(2))) float v2f;
typedef __attribute__((ext_vector_type(4))) float v4f;
typedef __attribute__((ext_vector_type(8))) float v8f;

#define B_  64
#define I_  2048
#define P_  8
#define J_  32
#define D_  16
#define KS_ 16              // K-dim split for the WMMA GEMM
#define IPS_ (I_ / KS_)     // i's per K slice = 128

// ---------------- ws layout (floats) ----------------
// Wb : [I][J][kh=2][n=16][4]   B-operand packed per-lane (8,388,608 f)
// Wsum: [I][J][P]              (524,288 f)
// xA : [I][kh=2][b=64][4]      A-operand packed per-lane (1,048,576 f)
// c2 : [I][J][B]               (4,194,304 f)
// s  : [B][J][D]               (32,768 f)
#define WB_OFF    ((size_t)0)
#define WSUM_OFF  (WB_OFF   + (size_t)I_*J_*2*D_*4)
#define XA_OFF    (WSUM_OFF + (size_t)I_*J_*P_)
#define C2_OFF    (XA_OFF   + (size_t)I_*2*B_*4)
#define S_OFF     (C2_OFF   + (size_t)I_*J_*B_)

// q -> p map shared by the packers and the GEMM:
//   q=0 -> p=kh*2, q=1 -> p=kh*2+1, q=2 -> p=4+kh*2, q=3 -> p=4+kh*2+1
// so WMMA step ph uses (q=2*ph, q=2*ph+1) == K = ph*4 + kh*2 + {0,1}.

// 1) Single pass over W: emit Wb (WMMA-lane-packed) and Wsum = sum_d W.
__global__ void prep_w_kernel(const float* __restrict__ W,
                              float* __restrict__ Wb,
                              float* __restrict__ Wsum) {
    int t = blockIdx.x * blockDim.x + threadIdx.x;   // (i*J + j)
    if (t >= I_ * J_) return;
    const float* wp = W + (size_t)t * (D_ * P_);     // W[i][j][d][p]
    float* wb = Wb + (size_t)t * (2 * D_ * 4);       // Wb[i][j][kh][n][4]
    float acc[P_];
#pragma unroll
    for (int p = 0; p < P_; ++p) acc[p] = 0.0f;
#pragma unroll
    for (int d = 0; d < D_; ++d) {
        float w[P_];
#pragma unroll
        for (int p = 0; p < P_; ++p) { w[p] = wp[d * P_ + p]; acc[p] += w[p]; }
        v4f lo = { w[0], w[1], w[4], w[5] };         // kh = 0
        v4f hi = { w[2], w[3], w[6], w[7] };         // kh = 1
        *(v4f*)(wb + ((size_t)(0 * D_ + d)) * 4) = lo;
        *(v4f*)(wb + ((size_t)(1 * D_ + d)) * 4) = hi;
    }
    float* op = Wsum + (size_t)t * P_;
#pragma unroll
    for (int p = 0; p < P_; ++p) op[p] = acc[p];
}

// 2) Pack x into the A-operand layout xA[i][kh][b][4].
__global__ void xpack_kernel(const float* __restrict__ x, float* __restrict__ xA) {
    int t = blockIdx.x * blockDim.x + threadIdx.x;   // b*I + i
    if (t >= B_ * I_) return;
    int b = t / I_, i = t % I_;
    const float* xp = x + (size_t)t * P_;
    float v[P_];
#pragma unroll
    for (int p = 0; p < P_; ++p) v[p] = xp[p];
    v4f lo = { v[0], v[1], v[4], v[5] };             // kh = 0
    v4f hi = { v[2], v[3], v[6], v[7] };             // kh = 1
    *(v4f*)(xA + ((size_t)(i * 2 + 0) * B_ + b) * 4) = lo;
    *(v4f*)(xA + ((size_t)(i * 2 + 1) * B_ + b) * 4) = hi;
}

// 3) Routing collapsed to closed form for the final coupling coefficients:
//    uhs = x . Wsum ; b1 = uhs/32 ; c1 = softmax(b1) ; b2 = b1 + c1*uhs ;
//    c2 = softmax(b2).  One thread per (b,i); Wsum[i] staged in LDS and
//    shared by the 64 batch threads of the block.
__global__ void routing_kernel(const float* __restrict__ x,
                               const float* __restrict__ Wsum,
                               float* __restrict__ c2) {
    __shared__ float lws[J_ * P_];                   // 1 KB
    int i = blockIdx.x;                              // 0..I_-1
    int b = threadIdx.x;                             // 0..63
    for (int k = threadIdx.x; k < J_ * P_; k += blockDim.x)
        lws[k] = Wsum[(size_t)i * (J_ * P_) + k];
    __syncthreads();

    float xv[P_];
    const float* xp = x + ((size_t)b * I_ + i) * P_;
#pragma unroll
    for (int p = 0; p < P_; ++p) xv[p] = xp[p];

    float uhs[J_];
#pragma unroll
    for (int j = 0; j < J_; ++j) {
        float s = 0.0f;
#pragma unroll
        for (int p = 0; p < P_; ++p) s += xv[p] * lws[j * P_ + p];
        uhs[j] = s;
    }

    float bb[J_], e[J_];
#pragma unroll
    for (int j = 0; j < J_; ++j) bb[j] = uhs[j] * (1.0f / (float)J_);

    // softmax #1 -> c1, fold b update
    float m = -3.4e38f;
#pragma unroll
    for (int j = 0; j < J_; ++j) m = fmaxf(m, bb[j]);
    float sum = 0.0f;
#pragma unroll
    for (int j = 0; j < J_; ++j) { e[j] = __expf(bb[j] - m); sum += e[j]; }
    float inv = 1.0f / sum;
#pragma unroll
    for (int j = 0; j < J_; ++j) bb[j] += (e[j] * inv) * uhs[j];

    // softmax #2 -> c2 (final coupling)
    m = -3.4e38f;
#pragma unroll
    for (int j = 0; j < J_; ++j) m = fmaxf(m, bb[j]);
    sum = 0.0f;
#pragma unroll
    for (int j = 0; j < J_; ++j) { e[j] = __expf(bb[j] - m); sum += e[j]; }
    inv = 1.0f / sum;

#pragma unroll
    for (int j = 0; j < J_; ++j)
        c2[((size_t)i * J_ + j) * B_ + b] = e[j] * inv;
}

// 4) zero the s accumulator (atomically accumulated by the GEMM)
__global__ void zero_kernel(float* __restrict__ s, int n) {
    int t = blockIdx.x * blockDim.x + threadIdx.x;
    if (t < n) s[t] = 0.0f;
}

// 5) Main contraction on the f32 tensor core:
//    s[b,j,d] = sum_{i,p} (c2[i,j,b]*x[b,i,p]) * W[i,j,d,p]
//    One wave per (j, batch-tile, K-slice). Per i: one b32 (c2) + two b128
//    (packed A/B operands) loads feed two chained v_wmma_f32_16x16x4_f32.
__global__ void gemm_kernel(const float* __restrict__ xA,
                            const float* __restrict__ c2,
                            const float* __restrict__ Wb,
                            float* __restrict__ s) {
    int bid   = blockIdx.x;
    int j     = bid % J_;
    int btile = (bid / J_) % (B_ / 16);
    int ks    = bid / (J_ * (B_ / 16));
    int i0    = ks * IPS_;

    int lane = threadIdx.x;          // 0..31, one full wave (EXEC all 1s)
    int m    = lane & 15;            // row within 16x16 tile (batch)
    int kh   = lane >> 4;            // lane-half selects K pair
    int b    = btile * 16 + m;
    int n    = lane & 15;            // column = d

    const float* xbase = xA + ((size_t)kh * B_ + b) * 4;                 // + i*2*B_*4
    const float* wbase = Wb + (((size_t)j * 2 + kh) * D_ + n) * 4;       // + i*J_*2*D_*4
    const float* cbase = c2 + (size_t)j * B_ + b;                        // + i*J_*B_

    v8f acc = {};
    for (int i = i0; i < i0 + IPS_; ++i) {
        float cv = cbase[(size_t)i * J_ * B_];
        v4f  xv = *(const v4f*)(xbase + (size_t)i * 2 * B_ * 4);
        v4f  wv = *(const v4f*)(wbase + (size_t)i * J_ * 2 * D_ * 4);

        v2f a0, b0, a1, b1;
        a0.x = cv * xv.x;  a0.y = cv * xv.y;         // WMMA step ph=0 (K=kh*2+{0,1})
        b0.x = wv.x;       b0.y = wv.y;
        acc = __builtin_amdgcn_wmma_f32_16x16x4_f32(
            false, a0, false, b0, (short)0, acc, false, false);

        a1.x = cv * xv.z;  a1.y = cv * xv.w;         // WMMA step ph=1 (K=4+kh*2+{0,1})
        b1.x = wv.z;       b1.y = wv.w;
        acc = __builtin_amdgcn_wmma_f32_16x16x4_f32(
            false, a1, false, b1, (short)0, acc, false, false);
    }

    // C/D layout: VGPR r -> M = r + 8*(lane>=16), N = lane&15
    int mbase = kh * 8;
#pragma unroll
    for (int r = 0; r < 8; ++r) {
        int bo = btile * 16 + mbase + r;
        atomicAdd(&s[((size_t)bo * J_ + j) * D_ + n], acc[r]);
    }
}

// 6) squash(s) -> out  (fully overwrites d_out)
__global__ void squash_kernel(const float* __restrict__ s, float* __restrict__ out) {
    int t = blockIdx.x * blockDim.x + threadIdx.x;   // b*J + j
    if (t >= B_ * J_) return;
    const float* sp = s + (size_t)t * D_;
    float v[D_], s2 = 0.0f;
#pragma unroll
    for (int d = 0; d < D_; ++d) { v[d] = sp[d]; s2 += v[d] * v[d]; }
    float scale = (s2 / (1.0f + s2)) * rsqrtf(s2 + 1e-7f);
    float* op = out + (size_t)t * D_;
#pragma unroll
    for (int d = 0; d < D_; ++d) op[d] = scale * v[d];
}

extern "C" void kernel_launch(void* const* d_in, const int* in_sizes, int n_in,
                              void* d_out, int out_size, void* d_ws, size_t ws_size,
                              hipStream_t stream) {
    const float* x = (const float*)d_in[0];   // [B, I, P]
    const float* W = (const float*)d_in[1];   // [1, I, J, D, P]
    float* out = (float*)d_out;               // [B, J, D]

    float* ws   = (float*)d_ws;
    float* Wb   = ws + WB_OFF;
    float* Wsum = ws + WSUM_OFF;
    float* xA   = ws + XA_OFF;
    float* c2   = ws + C2_OFF;
    float* sbuf = ws + S_OFF;

    (void)in_sizes; (void)n_in; (void)out_size; (void)ws_size;

    prep_w_kernel<<<(I_ * J_ + 255) / 256, 256, 0, stream>>>(W, Wb, Wsum);
    xpack_kernel<<<(B_ * I_ + 255) / 256, 256, 0, stream>>>(x, xA);
    routing_kernel<<<I_, B_, 0, stream>>>(x, Wsum, c2);
    zero_kernel<<<(B_ * J_ * D_ + 255) / 256, 256, 0, stream>>>(sbuf, B_ * J_ * D_);
    gemm_kernel<<<J_ * (B_ / 16) * KS_, 32, 0, stream>>>(xA, c2, Wb, sbuf);
    squash_kernel<<<(B_ * J_ + 255) / 256, 256, 0, stream>>>(sbuf, out);
}